// GPEncoder_58780922413929
// MI455X (gfx1250) — compile-verified
//
#include <hip/hip_runtime.h>
#include <math.h>

typedef __attribute__((ext_vector_type(2))) float v2f;
typedef __attribute__((ext_vector_type(8))) float v8f;

// explicit LDS (addrspace 3) pointers: 32-bit DS addressing, no generic casts
typedef __attribute__((address_space(3))) float ldsf;

#define LDP 68      // row stride 272B = 17*16: b128-aligned rows, 4-bank row skew
#define NTH 128     // 4 waves of 32

// ================= WMMA f32 16x16 tile primitives (LDS, stride LDP) ===========
// A operand (16x4 k-slab): VGPR v <-> K = kb*4 + v + 2*half, M = ln
__device__ __forceinline__ v2f lda_n(const ldsf* s, int r0, int c0, int kb, int half, int ln) {
  int ks = kb*4 + 2*half;
  v2f a; a[0] = s[(r0+ln)*LDP + c0 + ks]; a[1] = s[(r0+ln)*LDP + c0 + ks + 1]; return a;
}
__device__ __forceinline__ v2f lda_t(const ldsf* s, int r0, int c0, int kb, int half, int ln) {
  int ks = kb*4 + 2*half;
  v2f a; a[0] = s[(r0+ks)*LDP + c0 + ln]; a[1] = s[(r0+ks+1)*LDP + c0 + ln]; return a;
}
// B operand (4x16 k-slab): VGPR v <-> K = kb*4 + v + 2*half, N = ln
__device__ __forceinline__ v2f ldb_n(const ldsf* s, int r0, int c0, int kb, int half, int ln) {
  int ks = kb*4 + 2*half;
  v2f b; b[0] = s[(r0+ks)*LDP + c0 + ln]; b[1] = s[(r0+ks+1)*LDP + c0 + ln]; return b;
}
__device__ __forceinline__ v2f ldb_t(const ldsf* s, int r0, int c0, int kb, int half, int ln) {
  int ks = kb*4 + 2*half;
  v2f b; b[0] = s[(r0+ln)*LDP + c0 + ks]; b[1] = s[(r0+ln)*LDP + c0 + ks + 1]; return b;
}

// acc += op(A)(16 x 4*NKB) * op(B)(4*NKB x 16)
template<int NKB, bool TA, bool TB>
__device__ __forceinline__ v8f mmblk(const ldsf* As, int ar, int ac,
                                     const ldsf* Bs, int br, int bc,
                                     v8f acc, int half, int ln) {
  #pragma unroll
  for (int kb = 0; kb < NKB; ++kb) {
    v2f a = TA ? lda_t(As, ar, ac, kb, half, ln) : lda_n(As, ar, ac, kb, half, ln);
    v2f b = TB ? ldb_t(Bs, br, bc, kb, half, ln) : ldb_n(Bs, br, bc, kb, half, ln);
    acc = __builtin_amdgcn_wmma_f32_16x16x4_f32(false, a, false, b, (short)0, acc, false, false);
  }
  return acc;
}

// C/D layout: VGPR v -> row r0 + v + 8*half, col c0 + ln
__device__ __forceinline__ void store_c(ldsf* Cs, int r0, int c0, v8f acc, int half, int ln) {
  #pragma unroll
  for (int v = 0; v < 8; ++v) Cs[(r0 + v + 8*half)*LDP + c0 + ln] = acc[v];
}
__device__ __forceinline__ void sub_c(ldsf* Cs, int r0, int c0, v8f acc, int half, int ln) {
  #pragma unroll
  for (int v = 0; v < 8; ++v) Cs[(r0 + v + 8*half)*LDP + c0 + ln] -= acc[v];
}

// ---- async global->LDS copy of one 64x64 f32 tile (b128 per lane, ASYNCcnt) ----
__device__ __forceinline__ void async_tile_load(ldsf* Ms, const float* gp, int tid) {
  #pragma unroll
  for (int t = 0; t < 8; ++t) {
    int fi  = (tid + t*NTH) * 4;                 // fi % 4 == 0 -> 16B aligned
    int row = fi >> 6, col = fi & 63;
    const ldsf* dst = &Ms[row*LDP + col];        // row stride 272B -> 16B aligned
    const float* src = gp + fi;
    asm volatile("global_load_async_to_lds_b128 %0, %1, off"
                 :: "v"(dst), "v"(src) : "memory");
  }
}
__device__ __forceinline__ void wait_async0() {
  asm volatile("s_wait_asynccnt 0" ::: "memory");
}

// ============ blocked 64x64 Cholesky: M -> L (lower), Tinv[16 x 64] = all L_pp^-1 ===
__device__ __forceinline__ void chol_blocked64(ldsf* M, ldsf* Tinv, int tid) {
  const int lane = tid & 31, wave = tid >> 5, half = lane >> 4, ln = lane & 15;
  for (int p = 0; p < 4; ++p) {
    const int o = p*16;
    // 1) factor diagonal 16x16 block: wave-0 lanes do sqrt redundantly (lockstep),
    //    lane k stores the diagonal, lanes >k scale the column -> 2 barriers per col
    for (int k = 0; k < 16; ++k) {
      if (tid < 16) {
        float dk = sqrtf(M[(o+k)*LDP + o+k]);    // read happens before any store (lockstep)
        if (tid > k) M[(o+tid)*LDP + o+k] /= dk;
        if (tid == k) M[(o+k)*LDP + o+k] = dk;
      }
      __syncthreads();
      int rem = 15 - k;
      for (int idx = tid; idx < rem*rem; idx += NTH) {
        int i = k+1 + idx/rem, j = k+1 + idx%rem;
        if (j <= i) M[(o+i)*LDP + o+j] -= M[(o+i)*LDP + o+k]*M[(o+j)*LDP + o+k];
      }
      __syncthreads();
    }
    // 2) Tinv_p = L_pp^-1: thread c computes column c in registers (fully unrolled)
    if (tid < 16) {
      const int c = tid;
      float t[16];
      #pragma unroll
      for (int i = 0; i < 16; ++i) {
        float s = (i == c) ? 1.f : 0.f;
        #pragma unroll
        for (int j = 0; j < 16; ++j)
          if (j >= c && j < i) s -= M[(o+i)*LDP + o+j] * t[j];
        t[i] = (i < c) ? 0.f : s / M[(o+i)*LDP + o+i];
      }
      #pragma unroll
      for (int i = 0; i < 16; ++i) Tinv[i*LDP + o + c] = t[i];
    }
    __syncthreads();
    // 3) panel: L_ip = A_ip * Tinv_p^T (WMMA, wave-owned blocks)
    for (int i2 = p+1+wave; i2 < 4; i2 += 4) {
      v8f acc = {0.f,0.f,0.f,0.f,0.f,0.f,0.f,0.f};
      acc = mmblk<4,false,true>(M, i2*16, o, Tinv, 0, o, acc, half, ln);
      store_c(M, i2*16, o, acc, half, ln);
    }
    __syncthreads();
    // 4) trailing syrk: A_ij -= L_ip * L_jp^T (lower blocks, WMMA, round-robin)
    int cnt = 0;
    for (int i2 = p+1; i2 < 4; ++i2)
      for (int j2 = p+1; j2 <= i2; ++j2) {
        if ((cnt & 3) == wave) {
          v8f acc = {0.f,0.f,0.f,0.f,0.f,0.f,0.f,0.f};
          acc = mmblk<4,false,true>(M, i2*16, o, M, j2*16, o, acc, half, ln);
          sub_c(M, i2*16, j2*16, acc, half, ln);
        }
        ++cnt;
      }
    __syncthreads();
  }
}

// ====== in-place right TRSM: V := V * L^-T (V = 64x64, each wave owns a row strip) ==
// Row-block parallel: no cross-wave data flow -> no internal barriers needed.
__device__ __forceinline__ void trsm_right_lt64(const ldsf* M, const ldsf* Tinv,
                                                ldsf* V, int tid) {
  const int lane = tid & 31, wave = tid >> 5, half = lane >> 4, ln = lane & 15;
  const int r = wave*16;   // wave-owned 16-row strip
  for (int p = 0; p < 4; ++p) {
    const int o = p*16;
    if (p > 0) {
      v8f acc = {0.f,0.f,0.f,0.f,0.f,0.f,0.f,0.f};
      for (int q = 0; q < p; ++q)
        acc = mmblk<4,false,true>(V, r, q*16, M, o, q*16, acc, half, ln);  // V_rq * L_pq^T
      sub_c(V, r, o, acc, half, ln);
    }
    v8f acc2 = {0.f,0.f,0.f,0.f,0.f,0.f,0.f,0.f};
    acc2 = mmblk<4,false,true>(V, r, o, Tinv, 0, o, acc2, half, ln);       // U * Tinv_p^T
    store_c(V, r, o, acc2, half, ln);
  }
  __syncthreads();
}

// ========== solve (L L^T) v = v, blocked, using panel inverses ================
__device__ __forceinline__ void chol_solve_blocked64(const ldsf* M, const ldsf* Tinv,
                                                     ldsf* v, ldsf* stage, int tid) {
  for (int p = 0; p < 4; ++p) {            // forward: L y = v
    const int o = p*16;
    if (tid < 16) {
      float s = v[o + tid];
      for (int q = 0; q < p; ++q)
        #pragma unroll
        for (int j = 0; j < 16; ++j)
          s -= M[(o+tid)*LDP + q*16 + j] * v[q*16 + j];
      stage[tid] = s;
    }
    __syncthreads();
    if (tid < 16) {
      float s2 = 0.f;
      #pragma unroll
      for (int j = 0; j < 16; ++j) s2 += Tinv[tid*LDP + o + j] * stage[j];
      v[o + tid] = s2;
    }
    __syncthreads();
  }
  for (int p = 3; p >= 0; --p) {           // backward: L^T x = y
    const int o = p*16;
    if (tid < 16) {
      float s = v[o + tid];
      for (int q = p+1; q < 4; ++q)
        #pragma unroll
        for (int j = 0; j < 16; ++j)
          s -= M[(q*16+j)*LDP + o + tid] * v[q*16 + j];
      stage[tid] = s;
    }
    __syncthreads();
    if (tid < 16) {
      float s2 = 0.f;
      #pragma unroll
      for (int j = 0; j < 16; ++j) s2 += Tinv[j*LDP + o + tid] * stage[j];
      v[o + tid] = s2;
    }
    __syncthreads();
  }
}

__device__ __forceinline__ float reduce128(float val, ldsf* sred, int tid) {
  sred[tid] = val; __syncthreads();
  for (int s = 64; s > 0; s >>= 1) {
    if (tid < s) sred[tid] += sred[tid+s];
    __syncthreads();
  }
  float r = sred[0]; __syncthreads();
  return r;
}

__global__ __launch_bounds__(NTH) void gp_site_kernel(
    const float* __restrict__ x, const float* __restrict__ m,
    const float* __restrict__ R, const float* __restrict__ D,
    const float* __restrict__ S, float* __restrict__ out,
    float* __restrict__ p_ws)
{
  constexpr int L = 3, B = 1, C = 63, H = 64, W = 64, K = 64;
  constexpr int HW = H*W, KK = K*K;

  __shared__ float sh_mA[64*LDP];     // D -> L_D, later S -> L_S
  __shared__ float sh_mV[64*LDP];     // R -> V = R L_D^-T (in-place right TRSM)
  __shared__ float sh_mS[64*LDP];     // S1 = V V^T -> chol(S1~)
  __shared__ float sh_Ti[16*LDP];     // four 16x16 panel inverses, side by side
  __shared__ float sh_zv[64];
  __shared__ float sh_dv[64];
  __shared__ float sh_sv[64];
  __shared__ float sh_red[NTH];

  ldsf* mA   = (ldsf*)sh_mA;
  ldsf* mV   = (ldsf*)sh_mV;
  ldsf* mS   = (ldsf*)sh_mS;
  ldsf* Tinv = (ldsf*)sh_Ti;
  ldsf* zv   = (ldsf*)sh_zv;
  ldsf* dv   = (ldsf*)sh_dv;
  ldsf* sv   = (ldsf*)sh_sv;
  ldsf* sred = (ldsf*)sh_red;

  const int tid  = threadIdx.x;
  const int lane = tid & 31, wave = tid >> 5, half = lane >> 4, ln = lane & 15;
  const int site = blockIdx.x;                 // (l,b,h,w) flattened
  const int w = site % W;
  const int h = (site / W) % H;
  const int b = (site / HW) % B;
  const int l = site / (HW * B);
  const long base = (long)site * KK;

  // ---- async stream D and R straight into LDS (ASYNCcnt) ----
  async_tile_load(mA, D + base, tid);
  async_tile_load(mV, R + base, tid);
  wait_async0();
  __syncthreads();

  // ---- D = L L^T (blocked, WMMA panel + trailing updates) ----
  chol_blocked64(mA, Tinv, tid);

  // ---- V := R * L_D^-T (in-place, barrier-free right TRSM, all-WMMA) ----
  trsm_right_lt64(mA, Tinv, mV, tid);

  // ---- L_D dead: async stream S into mA; latency hidden behind Gram + Sxx chol ----
  async_tile_load(mA, S + base, tid);

  // ---- S1 = V V^T (WMMA Gram), into mS ----
  {
    const int bi = wave;
    #pragma unroll
    for (int bj = 0; bj < 4; ++bj) {
      v8f acc = {0.f,0.f,0.f,0.f,0.f,0.f,0.f,0.f};
      acc = mmblk<16,false,true>(mV, bi*16, 0, mV, bj*16, 0, acc, half, ln);
      store_c(mS, bi*16, bj*16, acc, half, ln);
    }
  }
  __syncthreads();

  // ---- Syx^T = col 63; then S1~ = blockdiag(Sxx, 1) so blocked-64 chol = chol(Sxx) ----
  if (tid < 63) zv[tid] = mS[tid*LDP + 63];
  if (tid == 0) zv[63] = 0.f;
  __syncthreads();
  if (tid < 63) mS[63*LDP + tid] = 0.f;
  if (tid == 0) mS[63*LDP + 63] = 1.f;
  __syncthreads();
  chol_blocked64(mS, Tinv, tid);
  chol_solve_blocked64(mS, Tinv, zv, sred, tid);   // zv[0:63] = Sxx^{-1} Syx^T

  // ---- xc, y ----
  const float* ml = m + (long)(l*B + b) * K;
  if (tid < 63) dv[tid] = x[((long)b*C + tid)*HW + h*W + w] - ml[tid];
  __syncthreads();
  float yp = (tid < 63) ? zv[tid]*dv[tid] : 0.f;
  float dot = reduce128(yp, sred, tid);            // y - my
  if (tid == 0) dv[63] = dot;
  __syncthreads();

  // ---- maha + logdet via blocked Cholesky of S (wait for async S first) ----
  wait_async0();
  __syncthreads();
  chol_blocked64(mA, Tinv, tid);
  if (tid < 64) sv[tid] = dv[tid];
  __syncthreads();
  chol_solve_blocked64(mA, Tinv, sv, sred, tid);
  float mp = (tid < 64) ? dv[tid]*sv[tid] : 0.f;
  float maha = reduce128(mp, sred, tid);
  float lp = (tid < 64) ? __logf(mA[tid*LDP+tid]) : 0.f;
  float logdet = 2.f * reduce128(lp, sred, tid);

  if (tid == 0) {
    const float LOG2PI = 1.8378770664093454f;
    float logp = -0.5f * (maha + logdet + (float)K * LOG2PI);
    const long YOFF = (long)B*L*C*HW;
    out[YOFF + ((long)(b*L + l))*HW + h*W + w] = ml[63] + dot;   // y_out (B,L,H,W)
    p_ws[((long)(l*B + b))*HW + h*W + w] = __expf(logp);         // per-l p
  }
}

// ---- tiling of x_out + product of p over l ----
__global__ __launch_bounds__(256) void gp_finalize_kernel(
    const float* __restrict__ x, const float* __restrict__ p_ws,
    float* __restrict__ out)
{
  constexpr int L = 3, B = 1, C = 63, H = 64, W = 64;
  constexpr int HW = H*W;
  const long XN   = (long)B*L*C*HW;
  const long POFF = XN + (long)B*L*HW;
  const long stride = (long)gridDim.x * blockDim.x;

  for (long idx = (long)blockIdx.x*blockDim.x + threadIdx.x; idx < XN; idx += stride) {
    long hw = idx % HW;
    long t  = idx / HW;
    int  c  = (int)(t % C); t /= C;
    int  bb = (int)(t / L);
    out[idx] = x[((long)bb*C + c)*HW + hw];
  }
  for (long idx = (long)blockIdx.x*blockDim.x + threadIdx.x; idx < (long)B*HW; idx += stride) {
    long hw = idx % HW;
    int  bb = (int)(idx / HW);
    float p = 1.f;
    #pragma unroll
    for (int l = 0; l < L; ++l) p *= p_ws[((long)(l*B + bb))*HW + hw];
    out[POFF + idx] = p;
  }
}

extern "C" void kernel_launch(void* const* d_in, const int* in_sizes, int n_in,
                              void* d_out, int out_size, void* d_ws, size_t ws_size,
                              hipStream_t stream) {
  const float* x = (const float*)d_in[0];
  const float* m = (const float*)d_in[1];
  const float* R = (const float*)d_in[2];
  const float* D = (const float*)d_in[3];
  const float* S = (const float*)d_in[4];
  float* out  = (float*)d_out;
  float* p_ws = (float*)d_ws;   // L*B*H*W floats = 48 KB scratch

  const int sites = 3 * 1 * 64 * 64;   // L*B*H*W
  gp_site_kernel<<<dim3(sites), dim3(NTH), 0, stream>>>(x, m, R, D, S, out, p_ws);
  gp_finalize_kernel<<<dim3(1024), dim3(256), 0, stream>>>(x, p_ws, out);
}